// GraphSAGE_2388001816783
// MI455X (gfx1250) — compile-verified
//
#include <hip/hip_runtime.h>

typedef __attribute__((ext_vector_type(2))) float v2f;
typedef __attribute__((ext_vector_type(8))) float v8f;

#define BN_EPS 1e-5f

// ---------------------------------------------------------------- utilities
__global__ void zero_f32(float* __restrict__ p, long n) {
  long i = (long)blockIdx.x * blockDim.x + threadIdx.x;
  long stride = (long)gridDim.x * blockDim.x;
  for (; i < n; i += stride) p[i] = 0.0f;
}

__global__ void degree_kernel(const int* __restrict__ dst, float* __restrict__ deg, int n_edges) {
  int e = blockIdx.x * blockDim.x + threadIdx.x;
  if (e < n_edges) atomicAdd(&deg[dst[e]], 1.0f);
}

// one wave (32 lanes) per edge; lane-strided columns -> coalesced atomics
template <int F>
__global__ void edge_agg(const float* __restrict__ H, const int* __restrict__ src,
                         const int* __restrict__ dst, float* __restrict__ Agg, int n_edges) {
  long gid = (long)blockIdx.x * blockDim.x + threadIdx.x;
  int e = (int)(gid >> 5);
  int lane = (int)(gid & 31);
  if (e >= n_edges) return;
  const float* hs = H + (long)src[e] * F;
  float* ad = Agg + (long)dst[e] * F;
#pragma unroll
  for (int c = 0; c < F / 32; ++c)
    atomicAdd(&ad[lane + 32 * c], hs[lane + 32 * c]);
}

// ---------------------------------------------------------------- fused SAGE GEMM
// Out[16 rows] = Hs @ Ws + (Agg/max(deg,1)) @ Wn + Bias, via V_WMMA_F32_16X16X4_F32.
// Block: 32 x (N/16) threads. A tiles staged in LDS (padded, conflict-free b64 reads).
template <int K, int N>
__global__ __launch_bounds__(32 * (N / 16))
void sage_gemm(const float* __restrict__ Hs, const float* __restrict__ Agg,
               const float* __restrict__ Deg, const float* __restrict__ Ws,
               const float* __restrict__ Wn, const float* __restrict__ Bias,
               float* __restrict__ Out) {
  constexpr int NT = N / 16;       // waves per block
  constexpr int PAD = K + 2;       // 130-float row stride: bank-conflict-free ds_load_b64
  __shared__ float sA[16 * PAD];   // self features,   16 x K
  __shared__ float sG[16 * PAD];   // neigh features,  16 x K (deg-normalized)

  const int tx = threadIdx.x;           // 0..31 (lane)
  const int ty = threadIdx.y;           // 0..NT-1 (N tile)
  const int tid = ty * 32 + tx;
  const int m0 = blockIdx.x * 16;

  // cooperative float4 staging of both A tiles
  constexpr int THREADS = 32 * NT;
  constexpr int VECS = 16 * (K / 4);
  const float4* Hs4 = (const float4*)(Hs + (long)m0 * K);
  const float4* Ag4 = (const float4*)(Agg + (long)m0 * K);
  for (int v = tid; v < VECS; v += THREADS) {
    int r = v / (K / 4);
    int c = (v % (K / 4)) * 4;
    float4 hv = Hs4[v];
    float4 av = Ag4[v];
    float invd = 1.0f / fmaxf(Deg[m0 + r], 1.0f);
    float* pa = &sA[r * PAD + c];
    pa[0] = hv.x; pa[1] = hv.y; pa[2] = hv.z; pa[3] = hv.w;
    float* pg = &sG[r * PAD + c];
    pg[0] = av.x * invd; pg[1] = av.y * invd; pg[2] = av.z * invd; pg[3] = av.w * invd;
  }
  __syncthreads();

  const int half = tx >> 4;     // ISA A/B layout: lanes<16 carry K {k,k+1}; lanes>=16 {k+2,k+3}
  const int r = tx & 15;        // M (for A) / N (for B,C,D) index within tile
  const int n = ty * 16 + r;

  float bv = Bias[n];
  v8f acc = {bv, bv, bv, bv, bv, bv, bv, bv};

  for (int k = 0; k < K; k += 4) {
    const int kk = k + half * 2;
    v2f a  = *(const v2f*)&sA[r * PAD + kk];
    v2f ag = *(const v2f*)&sG[r * PAD + kk];
    v2f b  = { Ws[(long)kk * N + n], Ws[(long)(kk + 1) * N + n] };
    v2f bn = { Wn[(long)kk * N + n], Wn[(long)(kk + 1) * N + n] };
    acc = __builtin_amdgcn_wmma_f32_16x16x4_f32(false, a,  false, b,  (short)0, acc, false, false);
    acc = __builtin_amdgcn_wmma_f32_16x16x4_f32(false, ag, false, bn, (short)0, acc, false, false);
  }

  // D layout: VGPR j -> row j (lanes<16) / row j+8 (lanes>=16), col = lane%16
  float* out = Out + (long)m0 * N + n;
#pragma unroll
  for (int j = 0; j < 8; ++j)
    out[(long)(j + half * 8) * N] = acc[j];
}

// ---------------------------------------------------------------- batchnorm pieces
// stat layout (floats): [0..127]=colsum  [128..255]=colsumsq  [256..383]=scale  [384..511]=shift
template <int F>
__global__ void col_reduce(const float* __restrict__ H, float* __restrict__ stat, int n_rows) {
  int c = threadIdx.x;                      // blockDim.x == F
  int r0 = blockIdx.x * 512;
  int r1 = (r0 + 512 < n_rows) ? (r0 + 512) : n_rows;
  float s = 0.0f, q = 0.0f;
  for (int rr = r0; rr < r1; ++rr) {
    float v = H[(long)rr * F + c];
    s += v;
    q = fmaf(v, v, q);
  }
  atomicAdd(&stat[c], s);
  atomicAdd(&stat[128 + c], q);
}

__global__ void bn_params(float* __restrict__ stat, const float* __restrict__ gamma,
                          const float* __restrict__ beta, float inv_n, int F) {
  int c = threadIdx.x;
  if (c >= F) return;
  float mu = stat[c] * inv_n;
  float var = stat[128 + c] * inv_n - mu * mu;   // biased variance (matches reference)
  float sc = gamma[c] * rsqrtf(var + BN_EPS);
  stat[256 + c] = sc;
  stat[384 + c] = beta[c] - mu * sc;
}

template <int F, bool RELU>
__global__ void bn_apply(const float* __restrict__ H, const float* __restrict__ stat,
                         float* __restrict__ Out, long n_elems) {
  long i = (long)blockIdx.x * blockDim.x + threadIdx.x;
  long stride = (long)gridDim.x * blockDim.x;
  for (; i < n_elems; i += stride) {
    int c = (int)(i & (F - 1));
    float v = fmaf(H[i], stat[256 + c], stat[384 + c]);
    if (RELU) v = fmaxf(v, 0.0f);
    Out[i] = v;
  }
}

// ---------------------------------------------------------------- launcher
extern "C" void kernel_launch(void* const* d_in, const int* in_sizes, int n_in,
                              void* d_out, int out_size, void* d_ws, size_t ws_size,
                              hipStream_t stream) {
  const float* feat = (const float*)d_in[0];
  const int*   src  = (const int*)d_in[1];
  const int*   dst  = (const int*)d_in[2];
  const float* Ws1  = (const float*)d_in[3];
  const float* Wn1  = (const float*)d_in[4];
  const float* b1   = (const float*)d_in[5];
  const float* g1   = (const float*)d_in[6];
  const float* be1  = (const float*)d_in[7];
  const float* Ws2  = (const float*)d_in[8];
  const float* Wn2  = (const float*)d_in[9];
  const float* b2   = (const float*)d_in[10];
  const float* g2   = (const float*)d_in[11];
  const float* be2  = (const float*)d_in[12];

  const int n_nodes = in_sizes[0] / 128;   // 100000 (multiple of 16)
  const int n_edges = in_sizes[1];         // 1600000

  // scratch layout: [deg | stat(512) | agg(n*128) | h1(n*128) | h2(n*64)]
  float* ws   = (float*)d_ws;
  const long deg_pad = ((long)n_nodes + 511) / 512 * 512;
  float* deg  = ws;
  float* stat = ws + deg_pad;
  float* agg  = stat + 512;
  float* h1   = agg + (long)n_nodes * 128;
  float* h2   = h1 + (long)n_nodes * 128;
  float* out  = (float*)d_out;

  const unsigned edge_thr_blocks = (unsigned)(((long)n_edges * 32 + 255) / 256);
  const float inv_n = 1.0f / (float)n_nodes;

  // ---- layer 1
  zero_f32<<<2048, 256, 0, stream>>>(deg, deg_pad + 512 + (long)n_nodes * 128);
  degree_kernel<<<(n_edges + 255) / 256, 256, 0, stream>>>(dst, deg, n_edges);
  edge_agg<128><<<edge_thr_blocks, 256, 0, stream>>>(feat, src, dst, agg, n_edges);
  {
    dim3 blk(32, 8);
    sage_gemm<128, 128><<<n_nodes / 16, blk, 0, stream>>>(feat, agg, deg, Ws1, Wn1, b1, h1);
  }
  col_reduce<128><<<(n_nodes + 511) / 512, 128, 0, stream>>>(h1, stat, n_nodes);
  bn_params<<<1, 128, 0, stream>>>(stat, g1, be1, inv_n, 128);
  bn_apply<128, true><<<2048, 256, 0, stream>>>(h1, stat, h1, (long)n_nodes * 128);

  // ---- layer 2
  zero_f32<<<2048, 256, 0, stream>>>(stat, 512 + (long)n_nodes * 128);  // stat+agg contiguous
  edge_agg<128><<<edge_thr_blocks, 256, 0, stream>>>(h1, src, dst, agg, n_edges);
  {
    dim3 blk(32, 4);
    sage_gemm<128, 64><<<n_nodes / 16, blk, 0, stream>>>(h1, agg, deg, Ws2, Wn2, b2, h2);
  }
  col_reduce<64><<<(n_nodes + 511) / 512, 64, 0, stream>>>(h2, stat, n_nodes);
  bn_params<<<1, 64, 0, stream>>>(stat, g2, be2, inv_n, 64);
  bn_apply<64, false><<<2048, 256, 0, stream>>>(h2, stat, out, (long)n_nodes * 64);
}